// HeEmb_1786706395652
// MI455X (gfx1250) — compile-verified
//
#include <hip/hip_runtime.h>

typedef __attribute__((ext_vector_type(2))) float v2f;
typedef __attribute__((ext_vector_type(8))) float v8f;

#define Bb 32
#define Nn 862
#define Ii 512
#define Oo 128
#define Ee 10
#define KC 128
#define XS_STRIDE 132   // 128 + 4 pad: avoids 64-bank conflicts on A-tile row reads
#define WS_STRIDE 132

#if defined(__has_builtin)
#if __has_builtin(__builtin_amdgcn_global_load_async_to_lds_b128)
#define HAVE_ASYNC_LDS 1
#endif
#endif

// Async global->LDS builtin takes pointers to 16-byte int vectors:
//   (v4i addrspace(1)*, v4i addrspace(3)*, imm offset, imm cpol)
typedef int v4i __attribute__((vector_size(16)));
typedef __attribute__((address_space(1))) v4i gvec;
typedef __attribute__((address_space(3))) v4i lvec;

__global__ __launch_bounds__(256, 1)
void moe_emb_fused(const float* __restrict__ x,
                   const float* __restrict__ gate_w,
                   const float* __restrict__ experts,
                   const float* __restrict__ ebias,
                   float* __restrict__ out)
{
    __shared__ float xs[Bb * XS_STRIDE];   // x chunk   [32][132]
    __shared__ float ws[KC * WS_STRIDE];   // w chunk   [128][132] (row = k, col = o)

    const int n    = blockIdx.x;           // channel
    const int tid  = threadIdx.x;
    const int lane = tid & 31;
    const int wave = tid >> 5;

    // ---- softmax gates (10 elems, computed redundantly per thread; uniform) ----
    float g[Ee];
    {
        const float* gw = gate_w + n * Ee;
        float m = -1e30f;
        #pragma unroll
        for (int e = 0; e < Ee; ++e) { g[e] = gw[e]; m = fmaxf(m, g[e]); }
        float s = 0.f;
        #pragma unroll
        for (int e = 0; e < Ee; ++e) { g[e] = __expf(g[e] - m); s += g[e]; }
        const float inv = 1.f / s;
        #pragma unroll
        for (int e = 0; e < Ee; ++e) g[e] *= inv;
    }

    // WMMA lane roles (V_WMMA_F32_16X16X4_F32):
    //   A 16x4: lanes 0-15 hold K=0,1 ; lanes 16-31 hold K=2,3 ; M = lane%16
    //   B 4x16: lanes 0-15 hold K=0,1 ; lanes 16-31 hold K=2,3 ; N = lane%16
    const int row  = lane & 15;
    const int klo  = (lane >> 4) << 1;       // 0 or 2
    const int ocol = wave * 16 + row;        // this wave's O-tile column

    v8f c0 = {}; // rows 0..15 of B(=32)
    v8f c1 = {}; // rows 16..31

    for (int chunk = 0; chunk < Ii / KC; ++chunk) {
        const int kc = chunk * KC;
        __syncthreads();   // protect LDS from previous iteration's readers

        // ---- stage x chunk: 32 rows x 128 k, one b128 per lane x4 ----
        // Async DMA global->LDS (ASYNCcnt): no VGPR data movement; overlaps
        // with the expert-combine phase below.
        #pragma unroll
        for (int j = 0; j < 4; ++j) {
            const int q  = tid + j * 256;        // 0..1023
            const int b  = q >> 5;               // 0..31
            const int kk = (q & 31) << 2;        // 0..124
            const float* gp = x + ((size_t)b * Nn + n) * Ii + kc + kk;
#ifdef HAVE_ASYNC_LDS
            __builtin_amdgcn_global_load_async_to_lds_b128(
                (gvec*)gp, (lvec*)&xs[b * XS_STRIDE + kk],
                /*offset=*/0, /*cpol=*/0);
#else
            *(float4*)(&xs[b * XS_STRIDE + kk]) = *(const float4*)gp;
#endif
        }

        // ---- build w chunk: ws[k][o] = sum_e g[e] * experts[e][kc+k][o] ----
        // experts stay L2-resident (2.6MB, reused by all 862 blocks)
        #pragma unroll 2
        for (int j = 0; j < 16; ++j) {
            const int p  = tid + j * 256;        // 0..4095
            const int k  = p >> 5;               // 0..127
            const int o4 = (p & 31) << 2;        // 0..124
            float ax = 0.f, ay = 0.f, az = 0.f, aw = 0.f;
            const float* ep = experts + ((size_t)(kc + k)) * Oo + o4;
            #pragma unroll
            for (int e = 0; e < Ee; ++e) {
                const float4 w4 = *(const float4*)(ep + (size_t)e * Ii * Oo);
                ax += g[e] * w4.x; ay += g[e] * w4.y;
                az += g[e] * w4.z; aw += g[e] * w4.w;
            }
            float4 acc; acc.x = ax; acc.y = ay; acc.z = az; acc.w = aw;
            *(float4*)(&ws[k * WS_STRIDE + o4]) = acc;
        }

#ifdef HAVE_ASYNC_LDS
        // Drain the async global->LDS copies before publishing the tile;
        // the compiler's barrier handling does not cover ASYNCcnt.
#if __has_builtin(__builtin_amdgcn_s_wait_asynccnt)
        __builtin_amdgcn_s_wait_asynccnt(0);
#else
        asm volatile("s_wait_asynccnt 0" ::: "memory");
#endif
#endif
        __syncthreads();

        // ---- WMMA sweep over the chunk: K steps of 4 ----
        #pragma unroll 4
        for (int kb = 0; kb < KC; kb += 4) {
            const v2f a0 = *(const v2f*)(&xs[row * XS_STRIDE + kb + klo]);
            const v2f a1 = *(const v2f*)(&xs[(row + 16) * XS_STRIDE + kb + klo]);
            v2f bv;
            bv.x = ws[(kb + klo)     * WS_STRIDE + ocol];
            bv.y = ws[(kb + klo + 1) * WS_STRIDE + ocol];
            c0 = __builtin_amdgcn_wmma_f32_16x16x4_f32(false, a0, false, bv,
                                                       (short)0, c0, false, false);
            c1 = __builtin_amdgcn_wmma_f32_16x16x4_f32(false, a1, false, bv,
                                                       (short)0, c1, false, false);
        }
    }

    // ---- bias for this lane's output column ----
    float bias = 0.f;
    #pragma unroll
    for (int e = 0; e < Ee; ++e) bias += g[e] * ebias[e * Oo + ocol];

    // C/D layout: VGPR r -> row (lane<16 ? r : r+8), col = lane%16
    const int rbase = (lane >> 4) << 3;   // 0 or 8
    #pragma unroll
    for (int r = 0; r < 8; ++r) {
        const int b0 = rbase + r;
        out[((size_t)b0        * Nn + n) * Oo + ocol] = c0[r] + bias;
        out[((size_t)(b0 + 16) * Nn + n) * Oo + ocol] = c1[r] + bias;
    }
}

extern "C" void kernel_launch(void* const* d_in, const int* in_sizes, int n_in,
                              void* d_out, int out_size, void* d_ws, size_t ws_size,
                              hipStream_t stream) {
    const float* x  = (const float*)d_in[0];   // [32, 862, 512]
    const float* gw = (const float*)d_in[1];   // [862, 10]
    const float* ex = (const float*)d_in[2];   // [10, 512, 128]
    const float* eb = (const float*)d_in[3];   // [10, 128]
    float* out = (float*)d_out;                // [32, 862, 128]

    dim3 grid(Nn), block(256);
    hipLaunchKernelGGL(moe_emb_fused, grid, block, 0, stream, x, gw, ex, eb, out);
}